// Codebook_20392504722120
// MI455X (gfx1250) — compile-verified
//
#include <hip/hip_runtime.h>

typedef __attribute__((ext_vector_type(2))) float v2f;
typedef __attribute__((ext_vector_type(8))) float v8f;

#define NUMV  8192
#define DIM   64
#define NROWS 32768
#define NTILE (NUMV / 16)   // 512 code tiles

// ws layout (units of 4 bytes):
//   [0, 8192)                 w2 (float)
//   [8192, 8192+32768)        indices (int)
//   [40960, 40960+8192)       counts (uint)
//   [49152, 49152+128)        commitment partial sums (float)
#define WS_W2      0
#define WS_IDX     8192
#define WS_COUNTS  (8192 + 32768)
#define WS_PARTIAL (8192 + 32768 + 8192)

__global__ void __launch_bounds__(256) vq_prep(const float* __restrict__ W,
                                               float* __restrict__ w2,
                                               unsigned* __restrict__ counts) {
    int k = blockIdx.x * 256 + threadIdx.x;
    if (k < NUMV) {
        const float4* wr = (const float4*)(W + k * DIM);
        float s = 0.f;
        #pragma unroll
        for (int c = 0; c < DIM / 4; ++c) {
            float4 w = wr[c];
            s += w.x * w.x + w.y * w.y + w.z * w.z + w.w * w.w;
        }
        w2[k] = s;
        counts[k] = 0u;
    }
}

// ---------------------------------------------------------------------------
// Fused distance + argmin kernel.
// 4 waves/block; each wave owns TWO 16-row tiles (M=32) sharing one B operand.
// Code tiles (16 codes x 64 ch) are double-buffered in LDS and staged with
// CDNA5 async global->LDS copies (ASYNCcnt), overlapping the WMMA compute.
// ---------------------------------------------------------------------------

// Stage one 16x64 f32 tile: 128 threads x 2 async b128 = 256 float4.
// Low 32 bits of a generic LDS pointer == LDS byte offset (aperture in hi32).
#define STAGE_TILE(LDS, CB)                                                    \
    {                                                                          \
        _Pragma("unroll")                                                      \
        for (int i_ = 0; i_ < 2; ++i_) {                                       \
            int q_    = (int)threadIdx.x + i_ * 128;                           \
            int code_ = q_ >> 4;                                               \
            int ch_   = (q_ & 15) * 4;                                         \
            unsigned loff_ = (unsigned)(size_t)((LDS) + code_ * 68 + ch_);     \
            const float* g_ = W + ((CB) + code_) * DIM + ch_;                  \
            asm volatile("global_load_async_to_lds_b128 %0, %1, off"           \
                         :: "v"(loff_), "v"(g_) : "memory");                   \
        }                                                                      \
    }

#define WAIT_ASYNC() asm volatile("s_wait_asynccnt 0" ::: "memory")

#define COMPUTE_TILE(LDS, CB)                                                  \
    {                                                                          \
        const int   code_ = (CB) + l16;                                        \
        const float w2v_  = w2[code_];                                         \
        v8f a0_ = {}, a1_ = {};                                                \
        _Pragma("unroll")                                                      \
        for (int k_ = 0; k_ < 16; ++k_) {                                      \
            v2f b_ = *(const v2f*)((LDS) + l16 * 68 + 4 * k_ + choff);         \
            a0_ = __builtin_amdgcn_wmma_f32_16x16x4_f32(                       \
                false, xa0[k_], false, b_, (short)0, a0_, false, false);       \
            a1_ = __builtin_amdgcn_wmma_f32_16x16x4_f32(                       \
                false, xa1[k_], false, b_, (short)0, a1_, false, false);       \
        }                                                                      \
        _Pragma("unroll")                                                      \
        for (int r_ = 0; r_ < 8; ++r_) {                                       \
            float s0_ = __builtin_fmaf(-2.0f, a0_[r_], w2v_);                  \
            float s1_ = __builtin_fmaf(-2.0f, a1_[r_], w2v_);                  \
            if (s0_ < minv0[r_]) { minv0[r_] = s0_; mini0[r_] = code_; }       \
            if (s1_ < minv1[r_]) { minv1[r_] = s1_; mini1[r_] = code_; }       \
        }                                                                      \
    }

__global__ void __launch_bounds__(128) vq_argmin(const float* __restrict__ x,
                                                 const float* __restrict__ W,
                                                 const float* __restrict__ w2,
                                                 int* __restrict__ indices) {
    __shared__ float tileW0[16 * 68];   // 68-float pitch -> conflict-free b64 reads
    __shared__ float tileW1[16 * 68];

    const int lane  = threadIdx.x & 31;
    const int wave  = threadIdx.x >> 5;
    const int half  = lane >> 4;        // 0: lanes 0-15, 1: lanes 16-31
    const int l16   = lane & 15;
    const int choff = half * 2;         // ISA f32 A/B layout: upper half holds K+2,K+3
    const int rowbase = (blockIdx.x * 4 + wave) * 32;

    // Kick off async stage of tile 0 before doing anything else.
    STAGE_TILE(tileW0, 0);

    // Preload A for both row tiles: xa[k] = x[row, 4k+choff .. 4k+choff+1]
    v2f xa0[16], xa1[16];
    {
        const int r0 = rowbase + l16;
        const int r1 = rowbase + 16 + l16;
        #pragma unroll
        for (int k = 0; k < 16; ++k) {
            xa0[k] = *(const v2f*)(x + r0 * DIM + 4 * k + choff);
            xa1[k] = *(const v2f*)(x + r1 * DIM + 4 * k + choff);
        }
    }

    float minv0[8], minv1[8];
    int   mini0[8], mini1[8];
    #pragma unroll
    for (int r = 0; r < 8; ++r) {
        minv0[r] = 3.0e38f; mini0[r] = 0;
        minv1[r] = 3.0e38f; mini1[r] = 0;
    }

    WAIT_ASYNC();
    __syncthreads();

    for (int t = 0; t < NTILE; t += 2) {
        // Stage t+1 into buf1 while computing tile t from buf0.
        STAGE_TILE(tileW1, (t + 1) * 16);
        COMPUTE_TILE(tileW0, t * 16);
        WAIT_ASYNC();
        __syncthreads();

        // Stage t+2 into buf0 while computing tile t+1 from buf1.
        if (t + 2 < NTILE) STAGE_TILE(tileW0, (t + 2) * 16);
        COMPUTE_TILE(tileW1, (t + 1) * 16);
        WAIT_ASYNC();
        __syncthreads();
    }

    // Per-row min+argmin across the 16 lane positions of each half-wave
    // (wave32: xor masks 1..8 stay within each 16-lane half).
    #pragma unroll
    for (int r = 0; r < 8; ++r) {
        float v0 = minv0[r]; int i0 = mini0[r];
        float v1 = minv1[r]; int i1 = mini1[r];
        #pragma unroll
        for (int m = 8; m >= 1; m >>= 1) {
            float ov0 = __shfl_xor(v0, m, 32); int oi0 = __shfl_xor(i0, m, 32);
            float ov1 = __shfl_xor(v1, m, 32); int oi1 = __shfl_xor(i1, m, 32);
            if (ov0 < v0 || (ov0 == v0 && oi0 < i0)) { v0 = ov0; i0 = oi0; }
            if (ov1 < v1 || (ov1 == v1 && oi1 < i1)) { v1 = ov1; i1 = oi1; }
        }
        if (l16 == 0) {
            indices[rowbase + half * 8 + r]      = i0;  // row tile 0
            indices[rowbase + 16 + half * 8 + r] = i1;  // row tile 1
        }
    }
}

__global__ void __launch_bounds__(256) vq_gather(const float* __restrict__ x,
                                                 const float* __restrict__ W,
                                                 const int* __restrict__ indices,
                                                 float* __restrict__ out,
                                                 unsigned* __restrict__ counts,
                                                 float* __restrict__ partial) {
    __shared__ float red[256];
    const int row = blockIdx.x * 256 + threadIdx.x;
    const int idx = indices[row];

    const float4* wr = (const float4*)(W + idx * DIM);
    const float4* xr = (const float4*)(x + row * DIM);
    float4*       od = (float4*)(out + row * DIM);

    float s = 0.f;
    #pragma unroll
    for (int c = 0; c < DIM / 4; ++c) {
        float4 w = wr[c], xv = xr[c];
        float a0 = w.x - xv.x, a1 = w.y - xv.y, a2 = w.z - xv.z, a3 = w.w - xv.w;
        s += a0 * a0 + a1 * a1 + a2 * a2 + a3 * a3;
        od[c] = w;   // forward value of straight-through estimator == z
    }
    atomicAdd(counts + idx, 1u);

    red[threadIdx.x] = s;
    __syncthreads();
    for (int st = 128; st > 0; st >>= 1) {
        if (threadIdx.x < st) red[threadIdx.x] += red[threadIdx.x + st];
        __syncthreads();
    }
    if (threadIdx.x == 0) partial[blockIdx.x] = red[0];
}

__global__ void __launch_bounds__(256) vq_finalize(const unsigned* __restrict__ counts,
                                                   const float* __restrict__ partial,
                                                   float* __restrict__ out) {
    __shared__ float red[256];
    float local = 0.f;
    for (int k = threadIdx.x; k < NUMV; k += 256) {
        float p = (float)counts[k] * (1.0f / (float)NROWS);
        local += p * logf(p + 1e-10f);
    }
    red[threadIdx.x] = local;
    __syncthreads();
    for (int st = 128; st > 0; st >>= 1) {
        if (threadIdx.x < st) red[threadIdx.x] += red[threadIdx.x + st];
        __syncthreads();
    }
    if (threadIdx.x == 0) {
        float cs = 0.f;
        for (int b = 0; b < 128; ++b) cs += partial[b];   // fixed order: deterministic
        out[NROWS * DIM + 0] = 0.0f;                        // codebook_loss
        out[NROWS * DIM + 1] = cs / (float)(NROWS * DIM);   // commitment_loss
        out[NROWS * DIM + 2] = expf(-red[0]);               // perplexity
    }
}

extern "C" void kernel_launch(void* const* d_in, const int* in_sizes, int n_in,
                              void* d_out, int out_size, void* d_ws, size_t ws_size,
                              hipStream_t stream) {
    const float* x = (const float*)d_in[0];   // [8,4096,64] f32
    const float* W = (const float*)d_in[1];   // [8192,64]   f32
    float*    out     = (float*)d_out;
    float*    ws      = (float*)d_ws;
    float*    w2      = ws + WS_W2;
    int*      indices = (int*)(ws + WS_IDX);
    unsigned* counts  = (unsigned*)(ws + WS_COUNTS);
    float*    partial = ws + WS_PARTIAL;

    vq_prep    <<<NUMV / 256, 256, 0, stream>>>(W, w2, counts);
    vq_argmin  <<<NROWS / 128, 128, 0, stream>>>(x, W, w2, indices);
    vq_gather  <<<NROWS / 256, 256, 0, stream>>>(x, W, indices, out, counts, partial);
    vq_finalize<<<1, 256, 0, stream>>>(counts, partial, out);
}